// ClusterPoolingNewCOOLayer_86028194939138
// MI455X (gfx1250) — compile-verified
//
#include <hip/hip_runtime.h>
#include <stdint.h>

// Problem constants (from the reference module).
#define NCLUSTERS 100000
#define SEG_CAP   4096          // per-bucket capacity (expected ~200, 6-sigma ~270)
#define SCAN_BLK  1024

// ---------------------------------------------------------------------------
// CDNA5 async global->LDS staging (gfx1250). Guarded so compile never breaks.
// clang-22 signature: (global int*, local int*, imm offset, imm cpol)
// ---------------------------------------------------------------------------
#if defined(__gfx1250__) && __has_builtin(__builtin_amdgcn_global_load_async_to_lds_b32)
#define USE_ASYNC_LDS 1
typedef __attribute__((address_space(1))) int gas_i32;
typedef __attribute__((address_space(3))) int las_i32;
#else
#define USE_ASYNC_LDS 0
#endif

static __device__ __forceinline__ void wait_async_zero() {
#if defined(__gfx1250__)
#if __has_builtin(__builtin_amdgcn_s_wait_asynccnt)
  __builtin_amdgcn_s_wait_asynccnt(0);
#else
  asm volatile("s_wait_asynccnt 0" ::: "memory");
#endif
#endif
}

// ---------------------------------------------------------------------------
// Kernel 0: zero the rec-cluster histogram (must be re-zeroed every replay).
// ---------------------------------------------------------------------------
__global__ void zero_u32_kernel(uint32_t* __restrict__ p, int n) {
  int i = blockIdx.x * blockDim.x + threadIdx.x;
  if (i < n) p[i] = 0u;
}

// ---------------------------------------------------------------------------
// Kernel 1: per-rec-cluster valid-edge histogram.
// ---------------------------------------------------------------------------
__global__ void count_kernel(const int* __restrict__ rec, const int* __restrict__ sen,
                             const int* __restrict__ clus, uint32_t* __restrict__ hist,
                             int E) {
  int i = blockIdx.x * blockDim.x + threadIdx.x;
  if (i >= E) return;
  // gfx1250: global_prefetch_b8, streaming hint
  __builtin_prefetch(rec + i + 8192, 0, 0);
  __builtin_prefetch(sen + i + 8192, 0, 0);
  int cr = clus[rec[i]];
  int cs = clus[sen[i]];
  if (cr >= 0 && cs >= 0) atomicAdd(&hist[cr], 1u);
}

// ---------------------------------------------------------------------------
// Kernel 2/5: single-block exclusive scan over n counters (n ~ 100k).
// out[n] = total. out2 (optional) gets a second copy (scatter cursors).
// ---------------------------------------------------------------------------
__global__ void scan_exclusive_kernel(const uint32_t* __restrict__ in,
                                      uint32_t* __restrict__ out,
                                      uint32_t* __restrict__ out2, int n) {
  __shared__ uint32_t tmp[SCAN_BLK];
  const int tid = threadIdx.x;
  uint32_t carry = 0u;
  for (int base = 0; base < n; base += SCAN_BLK) {
    int i = base + tid;
    uint32_t v = (i < n) ? in[i] : 0u;
    tmp[tid] = v;
    __syncthreads();
    for (int off = 1; off < SCAN_BLK; off <<= 1) {
      uint32_t t = (tid >= off) ? tmp[tid - off] : 0u;
      __syncthreads();
      tmp[tid] += t;
      __syncthreads();
    }
    uint32_t excl = carry + tmp[tid] - v;
    if (i < n) {
      out[i] = excl;
      if (out2) out2[i] = excl;
    }
    carry += tmp[SCAN_BLK - 1];
    __syncthreads();
  }
  if (tid == 0) out[n] = carry;
}

// ---------------------------------------------------------------------------
// Kernel 3: scatter sen-cluster of each valid edge into its rec-cluster segment.
// senBuf (80 MB) is L2-resident on MI455X (192 MB L2).
// ---------------------------------------------------------------------------
__global__ void scatter_kernel(const int* __restrict__ rec, const int* __restrict__ sen,
                               const int* __restrict__ clus, uint32_t* __restrict__ cursor,
                               uint32_t* __restrict__ senBuf, int E) {
  int i = blockIdx.x * blockDim.x + threadIdx.x;
  if (i >= E) return;
  __builtin_prefetch(rec + i + 8192, 0, 0);
  __builtin_prefetch(sen + i + 8192, 0, 0);
  int cr = clus[rec[i]];
  int cs = clus[sen[i]];
  if (cr >= 0 && cs >= 0) {
    uint32_t p = atomicAdd(&cursor[cr], 1u);
    senBuf[p] = (uint32_t)cs;
  }
}

// ---------------------------------------------------------------------------
// Kernel 4: per-bucket sort + dedup. One 256-thread (8-wave) block per bucket.
// Segment is staged into LDS via async global->LDS loads, bitonic sorted,
// deduped, written back compacted; unique count recorded.
// ---------------------------------------------------------------------------
__global__ void sort_dedup_kernel(uint32_t* __restrict__ senBuf,
                                  const uint32_t* __restrict__ segStart,
                                  uint32_t* __restrict__ uniqCnt) {
  __shared__ uint32_t vals[SEG_CAP];
  const int b = blockIdx.x;
  const uint32_t s0 = segStart[b];
  uint32_t n = segStart[b + 1] - s0;
  if (n > SEG_CAP) n = SEG_CAP;  // statistically unreachable (cap >> 6-sigma load)
  const uint32_t tid = threadIdx.x;
  if (n == 0) {
    if (tid == 0) uniqCnt[b] = 0u;
    return;
  }
  uint32_t M = 1;
  while (M < n) M <<= 1;

  // Stage segment into LDS (async DMA path on gfx1250).
  for (uint32_t j = tid; j < n; j += blockDim.x) {
#if USE_ASYNC_LDS
    __builtin_amdgcn_global_load_async_to_lds_b32(
        (gas_i32*)(senBuf + s0 + j), (las_i32*)&vals[j], 0, 0);
#else
    vals[j] = senBuf[s0 + j];
#endif
  }
  for (uint32_t j = n + tid; j < M; j += blockDim.x) vals[j] = 0xFFFFFFFFu;  // pad = +inf
#if USE_ASYNC_LDS
  wait_async_zero();
#endif
  __syncthreads();

  // Bitonic sort of M (power of two) elements, ascending.
  for (uint32_t k = 2; k <= M; k <<= 1) {
    for (uint32_t j = k >> 1; j > 0; j >>= 1) {
      for (uint32_t i = tid; i < M; i += blockDim.x) {
        uint32_t ij = i ^ j;
        if (ij > i) {
          uint32_t a = vals[i], c = vals[ij];
          bool up = ((i & k) == 0);
          if (up ? (a > c) : (a < c)) { vals[i] = c; vals[ij] = a; }
        }
      }
      __syncthreads();
    }
  }

  // Serial dedup (duplicates are ~1 in 10^6 edges; segment is ~200 long).
  if (tid == 0) {
    uint32_t u = 0, prev = 0xFFFFFFFFu;  // impossible value (ids < 2^17)
    for (uint32_t i = 0; i < n; ++i) {
      uint32_t v = vals[i];
      if (v != prev) { senBuf[s0 + u] = v; ++u; prev = v; }
    }
    uniqCnt[b] = u;
  }
}

// ---------------------------------------------------------------------------
// Kernel 6: fill output with -1 (pad slots).
// ---------------------------------------------------------------------------
__global__ void fill_kernel(int* __restrict__ out, int n) {
  int i = blockIdx.x * blockDim.x + threadIdx.x;
  if (i < n) out[i] = -1;
}

// ---------------------------------------------------------------------------
// Kernel 7: emit sorted unique (rec, sen) pairs compacted to the front.
// Bucket order == sorted-by-key order, so output is globally sorted.
// ---------------------------------------------------------------------------
__global__ void emit_kernel(const uint32_t* __restrict__ senBuf,
                            const uint32_t* __restrict__ segStart,
                            const uint32_t* __restrict__ uniqCnt,
                            const uint32_t* __restrict__ outOff,
                            int* __restrict__ outRec, int* __restrict__ outSen) {
  const int b = blockIdx.x;
  const uint32_t u = uniqCnt[b];
  const uint32_t s0 = segStart[b];
  const uint32_t o = outOff[b];
  for (uint32_t j = threadIdx.x; j < u; j += blockDim.x) {
    outRec[o + j] = b;
    outSen[o + j] = (int)senBuf[s0 + j];
  }
}

// ---------------------------------------------------------------------------
// Host launcher (graph-capture safe: stream-only, no mallocs/syncs).
// ---------------------------------------------------------------------------
extern "C" void kernel_launch(void* const* d_in, const int* in_sizes, int n_in,
                              void* d_out, int out_size, void* d_ws, size_t ws_size,
                              hipStream_t stream) {
  const int* rec  = (const int*)d_in[0];
  const int* sen  = (const int*)d_in[1];
  const int* clus = (const int*)d_in[2];
  const int E    = in_sizes[0];
  const int Eout = out_size / 2;  // receivers | senders, each Eout ints
  int* outRec = (int*)d_out;
  int* outSen = (int*)d_out + Eout;

  // Workspace partition (256B aligned slices).
  char* ws = (char*)d_ws;
  auto alignup = [](size_t x) { return (x + 255) & ~(size_t)255; };
  size_t off = 0;
  uint32_t* senBuf   = (uint32_t*)(ws + off); off += alignup((size_t)E * 4);
  uint32_t* hist     = (uint32_t*)(ws + off); off += alignup((size_t)(NCLUSTERS + 1) * 4);
  uint32_t* segStart = (uint32_t*)(ws + off); off += alignup((size_t)(NCLUSTERS + 1) * 4);
  uint32_t* cursor   = (uint32_t*)(ws + off); off += alignup((size_t)(NCLUSTERS + 1) * 4);
  uint32_t* uniqCnt  = (uint32_t*)(ws + off); off += alignup((size_t)(NCLUSTERS + 1) * 4);
  uint32_t* outOff   = (uint32_t*)(ws + off); off += alignup((size_t)(NCLUSTERS + 1) * 4);
  (void)ws_size; (void)n_in;

  const int T = 256;
  const int gridE = (E + T - 1) / T;

  // 0) zero histogram (re-done every replay for determinism)
  zero_u32_kernel<<<(NCLUSTERS + T - 1) / T, T, 0, stream>>>(hist, NCLUSTERS);
  // 1) histogram of rec clusters over valid edges
  count_kernel<<<gridE, T, 0, stream>>>(rec, sen, clus, hist, E);
  // 2) exclusive scan -> segment starts (+ cursor copy for scatter)
  scan_exclusive_kernel<<<1, SCAN_BLK, 0, stream>>>(hist, segStart, cursor, NCLUSTERS);
  // 3) scatter sen-cluster values into per-rec segments
  scatter_kernel<<<gridE, T, 0, stream>>>(rec, sen, clus, cursor, senBuf, E);
  // 4) per-bucket sort + dedup (async LDS staging, bitonic in LDS)
  sort_dedup_kernel<<<NCLUSTERS, T, 0, stream>>>(senBuf, segStart, uniqCnt);
  // 5) exclusive scan of unique counts -> output offsets
  scan_exclusive_kernel<<<1, SCAN_BLK, 0, stream>>>(uniqCnt, outOff, nullptr, NCLUSTERS);
  // 6) pad whole output with -1
  fill_kernel<<<(2 * Eout + T - 1) / T, T, 0, stream>>>((int*)d_out, 2 * Eout);
  // 7) emit sorted unique (rec, sen) pairs
  emit_kernel<<<NCLUSTERS, T, 0, stream>>>(senBuf, segStart, uniqCnt, outOff, outRec, outSen);
}